// HGTDetector_71090298683420
// MI455X (gfx1250) — compile-verified
//
#include <hip/hip_runtime.h>
#include <math.h>

// ---------------------------------------------------------------------------
// HGTConv (H=1, D=128) for MI455X / gfx1250, wave32 + WMMA f32 16x16x4 + TDM.
//
// Pipeline (all per-edge 128x128 matvecs hoisted to per-node GEMMs):
//   1) k_u = x_u@Wk+bk ; q_u = x_u@Wq+bq ; v_u = x_u@Wv+bv ; q_t = x_t@Wq+bq
//   2) kr_et = k_u @ a_rel[et] ; mr_et = v_u @ m_rel[et]   (et in {fo,fr,po})
//   3) per edge: ex = exp(SCALE*p_rel*dot(q[dst], kr[src])); denom[dst] += ex
//   4) per edge: msg[dst] += (ex/denom[dst]) * mr[src]      (f32 atomics)
//   5) out = beta*(gelu(msg)@Wa + ba) + (1-beta)*x,  beta = sigmoid(skip)
//
// Softmax max-subtraction dropped: shift-invariant, alphas bounded; empty
// segments stay exactly 0 because accumulators are zero-initialized.
// ---------------------------------------------------------------------------

#define DD 128
#define HGT_SCALE 0.08838834764831845f /* 1/sqrt(128) */

typedef float v2f __attribute__((ext_vector_type(2)));
typedef float v8f __attribute__((ext_vector_type(8)));
typedef unsigned int u32x4 __attribute__((ext_vector_type(4)));
typedef int i32x4 __attribute__((ext_vector_type(4)));
typedef int i32x8 __attribute__((ext_vector_type(8)));

// ---------------------------------------------------------------------------
// Y[n,128] = X[n,128] @ W[128,128] (+bias) (+gated skip blend)
//
// Block: 256 threads = 8 waves; macro-tile = 64 rows x 128 cols.
// Wave w owns cols [16w,16w+16) and 4 independent 16x16 accumulators
// (4 interleaved WMMA chains hide the D->C RAW latency; B-frags reused 4x).
//
// A tile staged into LDS by the Tensor Data Mover: 2D descriptor, 128-dword
// rows, pad_interval=6 (128 dw) + pad_amount=3 (4 dw) -> LDS row stride 132
// floats, which also kills the stride-128 bank conflict on column reads.
//
// The TDM's LDS write is invisible in IR, and an internal LDS global with no
// stores gets its loads folded to undef by GlobalOpt (round-3 asm), while
// volatile generic-pointer reads lower to serialized SCOPE_SYS flat loads
// (round-4 asm). Fix: one dynamically-never-executed, compiler-opaque store
// to As (n<0 guard) so the global is "stored-to", keeping plain addrspace(3)
// ds_loads alive; barrier + asm memory clobber pin them after the TDM wait.
//
// WMMA f32 16x16x4 VGPR layouts (ISA 7.12.2):
//   A 16x4 : lane L<16 -> row=L,  K={kb,kb+1}; lane>=16 -> row=L-16, K={kb+2,kb+3}
//   B 4x16 : lane L<16 -> col=L,  K rows {kb,kb+1}; lane>=16 -> col=L-16, {kb+2,kb+3}
//   C/D    : acc[r], lane L<16 -> (row r,   col L); lane>=16 -> (row r+8, col L-16)
// ---------------------------------------------------------------------------
__global__ __launch_bounds__(256) void gemm128_wmma(
    const float* __restrict__ X, const float* __restrict__ W,
    const float* __restrict__ bias, float* __restrict__ Y, int n,
    const float* __restrict__ skipX, const float* __restrict__ gate)
{
    __shared__ float As[64][132];   // LDS offset 0; row stride 132 = TDM pad

    const int row_base = blockIdx.x * 64;
    const int t = threadIdx.x;
    const int tile_rows = (n - row_base) < 64 ? (n - row_base) : 64;

    // Never executed at runtime (n > 0 always), but the compiler cannot prove
    // it: keeps GlobalOpt from folding As loads to undef (As has a store now).
    if (n < 0) As[0][t % 132] = (float)t;

#if defined(__has_builtin) && __has_builtin(__builtin_amdgcn_tensor_load_to_lds)
    // ---- TDM: async tensor tile load Global -> LDS (wave 0 only; TDM is a
    // per-wave op and ignores EXEC, so issuing from every wave would 8x it).
    if (t < 32) {
        unsigned long long ga =
            (unsigned long long)(const void*)(X + (size_t)row_base * DD);
        const unsigned td1 = 1u << 20;       // tensor_dim1: large, no OOB in play
        u32x4 g0;
        g0.x = 1u;                            // [1:0] count=1 (valid user D#)
        g0.y = 0u;                            // [63:32] lds_addr = 0 (As @ 0)
        g0.z = (unsigned)ga;                  // [95:64] global_addr lo
        g0.w = (unsigned)((ga >> 32) & 0x01FFFFFFu) | (2u << 30); // addr hi | type=2
        i32x8 g1;
        g1[0] = (2 << 16)      // data_size = 4B
              | (1 << 20)      // pad_enable
              | (6 << 22)      // pad_interval: 128 dwords
              | (3 << 25);     // pad_amount : 4 dwords  -> row stride 132 f32
        g1[1] = (int)((128u & 0xFFFFu) << 16);            // tensor_dim0=128 (lo16)
        g1[2] = (int)(((128u >> 16) & 0xFFFFu)            // tensor_dim0 hi
              | ((td1 & 0xFFFFu) << 16));                 // tensor_dim1 lo
        g1[3] = (int)(((td1 >> 16) & 0xFFFFu)             // tensor_dim1 hi
              | (128u << 16));                            // tile_dim0 = 128
        g1[4] = tile_rows & 0xFFFF;                       // tile_dim1; tile_dim2=0
        g1[5] = 128;                                      // tensor_dim0_stride lo
        g1[6] = 0;                                        // stride0 hi | stride1 lo
        g1[7] = 0;                                        // stride1 hi
        i32x4 zz4 = {0, 0, 0, 0};                         // 2D: groups 2/3 unused
        i32x8 zz8 = {0, 0, 0, 0, 0, 0, 0, 0};
        __builtin_amdgcn_tensor_load_to_lds(g0, g1, zz4, zz4, zz8, 0);
        __builtin_amdgcn_s_wait_tensorcnt(0);
    }
    asm volatile("" ::: "memory");   // keep LDS reads below the TDM wait
#else
    // Fallback: cooperative load (tile rows are contiguous in global).
    const float* Xt = X + (size_t)row_base * DD;
    for (int i = t; i < tile_rows * DD; i += 256)
        As[i >> 7][i & 127] = Xt[i];
#endif
    __syncthreads();

    const int lane = t & 31;
    const int wave = t >> 5;           // 0..7 -> column tile
    const int m    = lane & 15;
    const int kk   = (lane >> 4) * 2;  // 0 or 2
    const int col  = wave * 16 + m;

    v8f acc[4];
#pragma unroll
    for (int rt = 0; rt < 4; ++rt)
        acc[rt] = (v8f){0.f, 0.f, 0.f, 0.f, 0.f, 0.f, 0.f, 0.f};

#pragma unroll 4
    for (int kb = 0; kb < DD; kb += 4) {
        v2f b;
        b.x = W[(size_t)(kb + kk) * DD + col];
        b.y = W[(size_t)(kb + kk + 1) * DD + col];
#pragma unroll
        for (int rt = 0; rt < 4; ++rt) {
            v2f a;                     // 8B-aligned LDS read -> ds_load_b64
            a.x = As[rt * 16 + m][kb + kk];
            a.y = As[rt * 16 + m][kb + kk + 1];
            // (neg_a, A, neg_b, B, c_mod, C, reuse_a, reuse_b)
            acc[rt] = __builtin_amdgcn_wmma_f32_16x16x4_f32(
                false, a, false, b, (short)0, acc[rt], false, false);
        }
    }

    float beta = 1.0f, invb = 0.0f;
    if (gate) {
        float s = gate[0];
        beta = 1.0f / (1.0f + expf(-s));
        invb = 1.0f - beta;
    }
    const float bc   = bias ? bias[col] : 0.0f;
    const int   rofs = (lane >> 4) * 8;

#pragma unroll
    for (int rt = 0; rt < 4; ++rt) {
        const int rbase = row_base + rt * 16;
        if (rbase >= n) break;         // whole 16-row sub-tiles only (n % 16 == 0)
#pragma unroll
        for (int r = 0; r < 8; ++r) {
            size_t o = (size_t)(rbase + rofs + r) * DD + col;
            float y = acc[rt][r] + bc;
            if (skipX) y = beta * y + invb * skipX[o];
            Y[o] = y;
        }
    }
}

// ---------------------------------------------------------------------------
// Per-edge attention logits: one wave32 per edge.
// ex[e] = exp(SCALE * p_rel * dot(Q[dst], KR[src])) ; denom[dst] += ex[e]
// ---------------------------------------------------------------------------
__global__ __launch_bounds__(256) void edge_alpha(
    const float* __restrict__ Q, const float* __restrict__ KR,
    const int* __restrict__ ei, int E, const float* __restrict__ p_rel,
    float* __restrict__ ex, float* __restrict__ denom)
{
    const int wv = (int)((blockIdx.x * blockDim.x + threadIdx.x) >> 5);
    const int lane = threadIdx.x & 31;
    if (wv >= E) return;

    const int src = ei[wv];
    const int dst = ei[E + wv];

    const float4 qv = ((const float4*)(Q  + (size_t)dst * DD))[lane];
    const float4 kv = ((const float4*)(KR + (size_t)src * DD))[lane];
    float s = qv.x * kv.x + qv.y * kv.y + qv.z * kv.z + qv.w * kv.w;
#pragma unroll
    for (int off = 16; off > 0; off >>= 1) s += __shfl_xor(s, off, 32);

    if (lane == 0) {
        float e = expf(s * p_rel[0] * HGT_SCALE);
        ex[wv] = e;
        atomicAdd(&denom[dst], e);
    }
}

// ---------------------------------------------------------------------------
// Per-edge weighted scatter: msg[dst] += (ex[e]/denom[dst]) * MR[src]
// ---------------------------------------------------------------------------
__global__ __launch_bounds__(256) void edge_scatter(
    const float* __restrict__ MR, const int* __restrict__ ei, int E,
    const float* __restrict__ ex, const float* __restrict__ denom,
    float* __restrict__ msg)
{
    const int wv = (int)((blockIdx.x * blockDim.x + threadIdx.x) >> 5);
    const int lane = threadIdx.x & 31;
    if (wv >= E) return;

    const int src = ei[wv];
    const int dst = ei[E + wv];

    float w = 0.0f;
    if (lane == 0) w = ex[wv] / fmaxf(denom[dst], 1e-16f);
    w = __shfl(w, 0, 32);

    const float4 mv = ((const float4*)(MR + (size_t)src * DD))[lane];
    float* out = msg + (size_t)dst * DD + lane * 4;
    atomicAdd(out + 0, w * mv.x);
    atomicAdd(out + 1, w * mv.y);
    atomicAdd(out + 2, w * mv.z);
    atomicAdd(out + 3, w * mv.w);
}

// ---------------------------------------------------------------------------
// Exact GELU (matches jax.nn.gelu approximate=False), in place.
// ---------------------------------------------------------------------------
__global__ __launch_bounds__(256) void gelu_inplace(float* __restrict__ x, int n)
{
    int i = blockIdx.x * blockDim.x + threadIdx.x;
    if (i < n) {
        float v = x[i];
        x[i] = 0.5f * v * (1.0f + erff(v * 0.70710678118654752f));
    }
}

// ---------------------------------------------------------------------------
// Host side
// ---------------------------------------------------------------------------

// Input index map (setup_inputs() dict insertion order, nested depth-first):
enum {
    IN_XU = 0, IN_XT = 1,
    // params['user'] : Wk,bk,Wq,bq,Wv,bv,Wa,ba,skip
    IN_U_WK = 2, IN_U_BK, IN_U_WQ, IN_U_BQ, IN_U_WV, IN_U_BV, IN_U_WA, IN_U_BA, IN_U_SKIP,
    // params['tweet']
    IN_T_WK = 11, IN_T_BK, IN_T_WQ, IN_T_BQ, IN_T_WV, IN_T_BV, IN_T_WA, IN_T_BA, IN_T_SKIP,
    // params['follow'|'friend'|'post'] : a_rel, m_rel, p_rel
    IN_FO_A = 20, IN_FO_M, IN_FO_P,
    IN_FR_A = 23, IN_FR_M, IN_FR_P,
    IN_PO_A = 26, IN_PO_M, IN_PO_P,
    IN_EI_FO = 29, IN_EI_FR = 30, IN_EI_PO = 31
};

extern "C" void kernel_launch(void* const* d_in, const int* in_sizes, int n_in,
                              void* d_out, int out_size, void* d_ws, size_t ws_size,
                              hipStream_t stream)
{
    (void)n_in; (void)out_size; (void)ws_size;

    const int NU = in_sizes[IN_XU] / DD;   // 50000
    const int NT = in_sizes[IN_XT] / DD;   // 100000
    const int E_fo = in_sizes[IN_EI_FO] / 2;
    const int E_fr = in_sizes[IN_EI_FR] / 2;
    const int E_po = in_sizes[IN_EI_PO] / 2;

    const float* x_u = (const float*)d_in[IN_XU];
    const float* x_t = (const float*)d_in[IN_XT];
    const int* ei_fo = (const int*)d_in[IN_EI_FO];
    const int* ei_fr = (const int*)d_in[IN_EI_FR];
    const int* ei_po = (const int*)d_in[IN_EI_PO];

    // ---- workspace carve-out (floats) ----
    float* base = (float*)d_ws;
    size_t o = 0;
    float* k_u   = base + o; o += (size_t)NU * DD;
    float* q_u   = base + o; o += (size_t)NU * DD;
    float* v_u   = base + o; o += (size_t)NU * DD;
    float* q_t   = base + o; o += (size_t)NT * DD;
    float* kr_fo = base + o; o += (size_t)NU * DD;
    float* kr_fr = base + o; o += (size_t)NU * DD;
    float* kr_po = base + o; o += (size_t)NU * DD;
    float* mr_fo = base + o; o += (size_t)NU * DD;
    float* mr_fr = base + o; o += (size_t)NU * DD;
    float* mr_po = base + o; o += (size_t)NU * DD;
    // zero-initialized region (single memset): denoms + message accumulators
    float* zreg    = base + o;
    float* denom_u = base + o; o += NU;
    float* denom_t = base + o; o += NT;
    float* msg_u   = base + o; o += (size_t)NU * DD;
    float* msg_t   = base + o; o += (size_t)NT * DD;
    size_t zbytes = (size_t)(base + o - zreg) * sizeof(float);
    float* ex_fo = base + o; o += E_fo;
    float* ex_fr = base + o; o += E_fr;
    float* ex_po = base + o; o += E_po;

    float* out_u = (float*)d_out;
    float* out_t = out_u + (size_t)NU * DD;

    (void)hipMemsetAsync(zreg, 0, zbytes, stream);

    const dim3 blk(256);
    const int gU = (NU + 63) / 64;
    const int gT = (NT + 63) / 64;

    // 1) node projections (WMMA GEMMs, TDM-staged A tiles)
    gemm128_wmma<<<gU, blk, 0, stream>>>(x_u, (const float*)d_in[IN_U_WK], (const float*)d_in[IN_U_BK], k_u, NU, nullptr, nullptr);
    gemm128_wmma<<<gU, blk, 0, stream>>>(x_u, (const float*)d_in[IN_U_WQ], (const float*)d_in[IN_U_BQ], q_u, NU, nullptr, nullptr);
    gemm128_wmma<<<gU, blk, 0, stream>>>(x_u, (const float*)d_in[IN_U_WV], (const float*)d_in[IN_U_BV], v_u, NU, nullptr, nullptr);
    gemm128_wmma<<<gT, blk, 0, stream>>>(x_t, (const float*)d_in[IN_T_WQ], (const float*)d_in[IN_T_BQ], q_t, NT, nullptr, nullptr);

    // 2) per-(src,edge-type) relation transforms (hoisted from per-edge)
    gemm128_wmma<<<gU, blk, 0, stream>>>(k_u, (const float*)d_in[IN_FO_A], nullptr, kr_fo, NU, nullptr, nullptr);
    gemm128_wmma<<<gU, blk, 0, stream>>>(k_u, (const float*)d_in[IN_FR_A], nullptr, kr_fr, NU, nullptr, nullptr);
    gemm128_wmma<<<gU, blk, 0, stream>>>(k_u, (const float*)d_in[IN_PO_A], nullptr, kr_po, NU, nullptr, nullptr);
    gemm128_wmma<<<gU, blk, 0, stream>>>(v_u, (const float*)d_in[IN_FO_M], nullptr, mr_fo, NU, nullptr, nullptr);
    gemm128_wmma<<<gU, blk, 0, stream>>>(v_u, (const float*)d_in[IN_FR_M], nullptr, mr_fr, NU, nullptr, nullptr);
    gemm128_wmma<<<gU, blk, 0, stream>>>(v_u, (const float*)d_in[IN_PO_M], nullptr, mr_po, NU, nullptr, nullptr);

    // 3) attention logits + segment denominators (wave per edge)
    edge_alpha<<<(E_fo + 7) / 8, blk, 0, stream>>>(q_u, kr_fo, ei_fo, E_fo, (const float*)d_in[IN_FO_P], ex_fo, denom_u);
    edge_alpha<<<(E_fr + 7) / 8, blk, 0, stream>>>(q_u, kr_fr, ei_fr, E_fr, (const float*)d_in[IN_FR_P], ex_fr, denom_u);
    edge_alpha<<<(E_po + 7) / 8, blk, 0, stream>>>(q_t, kr_po, ei_po, E_po, (const float*)d_in[IN_PO_P], ex_po, denom_t);

    // 4) softmax-weighted message scatter
    edge_scatter<<<(E_fo + 7) / 8, blk, 0, stream>>>(mr_fo, ei_fo, E_fo, ex_fo, denom_u, msg_u);
    edge_scatter<<<(E_fr + 7) / 8, blk, 0, stream>>>(mr_fr, ei_fr, E_fr, ex_fr, denom_u, msg_u);
    edge_scatter<<<(E_po + 7) / 8, blk, 0, stream>>>(mr_po, ei_po, E_po, ex_po, denom_t, msg_t);

    // 5) gelu -> output projection with gated skip (fused epilogue)
    gelu_inplace<<<((NU * DD) + 255) / 256, blk, 0, stream>>>(msg_u, NU * DD);
    gelu_inplace<<<((NT * DD) + 255) / 256, blk, 0, stream>>>(msg_t, NT * DD);

    gemm128_wmma<<<gU, blk, 0, stream>>>(msg_u, (const float*)d_in[IN_U_WA], (const float*)d_in[IN_U_BA], out_u, NU, x_u, (const float*)d_in[IN_U_SKIP]);
    gemm128_wmma<<<gT, blk, 0, stream>>>(msg_t, (const float*)d_in[IN_T_WA], (const float*)d_in[IN_T_BA], out_t, NT, x_t, (const float*)d_in[IN_T_SKIP]);
}